// MultiHeadAttentionHybrid_65481071404560
// MI455X (gfx1250) — compile-verified
//
#include <hip/hip_runtime.h>

// MultiHeadAttentionHybrid for MI455X (gfx1250, wave32).
// B=4, S=2048, E=16, H=4, d_k=4.  Flash-attention with V_WMMA_F32_16X16X4_F32
// (exact fit for d_k=4, full fp32).  Score tensor (268 MB) never materialized.
// Total work ~1.1 GFLOP / ~1.5 MB HBM traffic -> latency bound; the win is
// keeping everything in registers/LDS and on the XDL (WMMA) pipe.

#define BB 4
#define SS 2048
#define EE 16
#define HH 4
#define QTILES_PER_B (SS / 16)   // 128
#define TOKENS (BB * SS)         // 8192

typedef __attribute__((ext_vector_type(2))) float v2f;
typedef __attribute__((ext_vector_type(8))) float v8f;

static __device__ __forceinline__ v8f wmma4(v2f a, v2f b, v8f c) {
  // D(16x16,f32) = A(16x4,f32) * B(4x16,f32) + C
  return __builtin_amdgcn_wmma_f32_16x16x4_f32(
      /*neg_a=*/false, a, /*neg_b=*/false, b,
      /*c_mod=*/(short)0, c, /*reuse_a=*/false, /*reuse_b=*/false);
}

// Row reductions across the 16 lanes of a half-wave (C/D column n=lane%16).
// xor masks 1,2,4,8 never cross the 16-lane half boundary.
static __device__ __forceinline__ float rowmax16(float v) {
  v = fmaxf(v, __shfl_xor(v, 1, 32));
  v = fmaxf(v, __shfl_xor(v, 2, 32));
  v = fmaxf(v, __shfl_xor(v, 4, 32));
  v = fmaxf(v, __shfl_xor(v, 8, 32));
  return v;
}
static __device__ __forceinline__ float rowsum16(float v) {
  v += __shfl_xor(v, 1, 32);
  v += __shfl_xor(v, 2, 32);
  v += __shfl_xor(v, 4, 32);
  v += __shfl_xor(v, 8, 32);
  return v;
}

// ---------------------------------------------------------------------------
// Kernel 1: K = X * Wk^T, V = X * Wv^T   ([8192,16] x [16,16], WMMA 16x16x4)
// One wave per 16-token tile; 512 tiles -> 64 blocks x 8 waves.
// ---------------------------------------------------------------------------
__global__ __launch_bounds__(256) void qkv_kernel(
    const float* __restrict__ x, const float* __restrict__ Wk,
    const float* __restrict__ Wv, float* __restrict__ Kb,
    float* __restrict__ Vb) {
  const int lane = threadIdx.x & 31;
  const int wave = threadIdx.x >> 5;
  const int lo = lane & 15;
  const int hi = lane >> 4;
  const int tile = blockIdx.x * 8 + wave;
  const int t0 = tile * 16;

  const float* xr = x + (size_t)(t0 + lo) * EE;
  v2f a[4], bk[4], bv[4];
#pragma unroll
  for (int kb = 0; kb < 4; ++kb) {
    const int c = kb * 4 + 2 * hi;               // global contraction index
    a[kb].x = xr[c];                             // A[m=lo][k=c]
    a[kb].y = xr[c + 1];
    bk[kb].x = Wk[lo * EE + c];                  // B[k][n=lo] = Wk[n][k]
    bk[kb].y = Wk[lo * EE + c + 1];
    bv[kb].x = Wv[lo * EE + c];
    bv[kb].y = Wv[lo * EE + c + 1];
  }
  v8f ck = {};
  v8f cv = {};
#pragma unroll
  for (int kb = 0; kb < 4; ++kb) {
    ck = wmma4(a[kb], bk[kb], ck);
    cv = wmma4(a[kb], bv[kb], cv);
  }
#pragma unroll
  for (int j = 0; j < 8; ++j) {                  // D[m=j+8*hi][n=lo]
    const size_t row = (size_t)(t0 + j + 8 * hi) * EE + lo;
    Kb[row] = ck[j];
    Vb[row] = cv[j];
  }
}

// ---------------------------------------------------------------------------
// Kernel 2: flash attention + fused output projection.
// One wave per (batch, 16-query tile); 4*128 = 512 waves -> 64 blocks x 8.
// Per 16-key tile: 4 WMMA (scores, all heads) + 16 WMMA (P*V, head-masked B
// columns accumulate into one unified 16x16 output fragment).
// ---------------------------------------------------------------------------
__global__ __launch_bounds__(256) void attn_kernel(
    const float* __restrict__ x, const float* __restrict__ theta,
    const float* __restrict__ Wc, const float* __restrict__ Kb,
    const float* __restrict__ Vb, float* __restrict__ out) {
  // Per-wave P staging: 4 heads x 16 rows x 17 (pad breaks bank conflicts).
  __shared__ float lds[8 * HH * 16 * 17];
  float* P = lds + (threadIdx.x >> 5) * (HH * 16 * 17);

  const int lane = threadIdx.x & 31;
  const int lo = lane & 15;
  const int hi = lane >> 4;
  const int myh = lo >> 2;                       // head owning output column lo
  const int tile = blockIdx.x * 8 + (threadIdx.x >> 5);
  const int b = tile >> 7;                       // / QTILES_PER_B
  const int q0 = (tile & (QTILES_PER_B - 1)) * 16;
  const int base = b * SS;                       // token base of this batch

  // Q fragments per head, pre-scaled by 1/sqrt(d_k) = 0.5.
  // Quantum head: <Z_i> = cos(x_i + theta_i).
  v2f aq[HH];
  const float* xq = x + (size_t)(base + q0 + lo) * EE;
#pragma unroll
  for (int h = 0; h < HH; ++h) {
    const int c = 4 * h + 2 * hi;
    aq[h].x = __cosf(xq[c] + theta[c]) * 0.5f;
    aq[h].y = __cosf(xq[c + 1] + theta[c + 1]) * 0.5f;
  }

  // Online-softmax state.  mrow: running max per (head, local row);
  // lsel: running denom for this lane's own output-column head.
  float mrow[HH][8];
  float lsel[8];
  v8f Oacc = {};
#pragma unroll
  for (int h = 0; h < HH; ++h)
#pragma unroll
    for (int j = 0; j < 8; ++j) mrow[h][j] = -3.0e38f;
#pragma unroll
  for (int j = 0; j < 8; ++j) lsel[j] = 0.0f;

  for (int kk = 0; kk < SS / 16; ++kk) {
    const int k0 = kk * 16;
    const float* krow = Kb + (size_t)(base + k0 + lo) * EE;  // B rows: n = lo

    // V rows for the 4 contraction chunks of P*V (all 16 embed cols).
    v2f vrow[4];
#pragma unroll
    for (int cc = 0; cc < 4; ++cc) {
      vrow[cc].x = Vb[(size_t)(base + k0 + 4 * cc + 2 * hi) * EE + lo];
      vrow[cc].y = Vb[(size_t)(base + k0 + 4 * cc + 2 * hi + 1) * EE + lo];
    }

    // ---- scores + online softmax, head by head ----
#pragma unroll
    for (int h = 0; h < HH; ++h) {
      v2f bkf;
      const int c = 4 * h + 2 * hi;
      bkf.x = krow[c];                           // B[k][n=key] = K[key][4h+k]
      bkf.y = krow[c + 1];
      v8f s = {};
      s = wmma4(aq[h], bkf, s);                  // S_h[16q x 16k], pre-scaled
#pragma unroll
      for (int j = 0; j < 8; ++j) {
        const float sv = s[j];
        const float rmax = rowmax16(sv);
        const float mold = mrow[h][j];
        const float mnew = fmaxf(mold, rmax);
        mrow[h][j] = mnew;
        const float alpha = __expf(mold - mnew);
        const float p = __expf(sv - mnew);
        const float rsum = rowsum16(p);
        if (myh == h) {                          // lane-select, no branch
          lsel[j] = lsel[j] * alpha + rsum;
          Oacc[j] *= alpha;
        }
        P[(h * 16 + j + 8 * hi) * 17 + lo] = p;  // stage P_h for transpose
      }
    }
    asm volatile("" ::: "memory");               // order LDS store -> load

    // ---- O += P_h * V_h via WMMA; B columns masked to head h ----
#pragma unroll
    for (int h = 0; h < HH; ++h) {
#pragma unroll
      for (int cc = 0; cc < 4; ++cc) {
        v2f ap;                                  // A[m=lo][k] = P_h[lo][4cc+k]
        ap.x = P[(h * 16 + lo) * 17 + 4 * cc + 2 * hi];
        ap.y = P[(h * 16 + lo) * 17 + 4 * cc + 2 * hi + 1];
        v2f bm;                                  // zero non-head-h columns
        bm.x = (myh == h) ? vrow[cc].x : 0.0f;
        bm.y = (myh == h) ? vrow[cc].y : 0.0f;
        Oacc = wmma4(ap, bm, Oacc);
      }
    }
    asm volatile("" ::: "memory");
  }

  // ---- normalize, then fused projection: out = O * Wc^T ----
#pragma unroll
  for (int j = 0; j < 8; ++j)
    P[(j + 8 * hi) * 17 + lo] = Oacc[j] / lsel[j];
  asm volatile("" ::: "memory");

  v8f dres = {};
#pragma unroll
  for (int kb = 0; kb < 4; ++kb) {
    const int c = kb * 4 + 2 * hi;
    v2f ao;
    ao.x = P[lo * 17 + c];
    ao.y = P[lo * 17 + c + 1];
    v2f bc;
    bc.x = Wc[lo * EE + c];                      // B[k][n=lo] = Wc[n][k]
    bc.y = Wc[lo * EE + c + 1];
    dres = wmma4(ao, bc, dres);
  }
#pragma unroll
  for (int j = 0; j < 8; ++j)
    out[(size_t)(base + q0 + j + 8 * hi) * EE + lo] = dres[j];
}

// ---------------------------------------------------------------------------
extern "C" void kernel_launch(void* const* d_in, const int* in_sizes, int n_in,
                              void* d_out, int out_size, void* d_ws,
                              size_t ws_size, hipStream_t stream) {
  (void)in_sizes; (void)n_in; (void)out_size; (void)ws_size;
  const float* x     = (const float*)d_in[0];   // [B,S,E]
  const float* theta = (const float*)d_in[1];   // [E]
  const float* Wk    = (const float*)d_in[2];   // [E,E]
  const float* Wv    = (const float*)d_in[3];   // [E,E]
  const float* Wc    = (const float*)d_in[4];   // [E,E]
  float* Kb = (float*)d_ws;                     // [B*S, E]
  float* Vb = Kb + (size_t)TOKENS * EE;         // [B*S, E]  (1 MB total)
  float* outp = (float*)d_out;

  qkv_kernel<<<TOKENS / 16 / 8, 256, 0, stream>>>(x, Wk, Wv, Kb, Vb);
  attn_kernel<<<(BB * QTILES_PER_B) / 8, 256, 0, stream>>>(x, theta, Wc, Kb,
                                                           Vb, outp);
}